// SSA_28845000360490
// MI455X (gfx1250) — compile-verified
//
#include <hip/hip_runtime.h>

// ---------------------------------------------------------------------------
// B=4, C=32, H=W=64, 2C=64 cat channels, 8 conv channels. All fp32.
//
// Algorithmic core: never materialize Projection [4096x4096]:
//   Y = V * ( (V^T V)^-1 * (V^T X1) )    (~0.2 GFLOP, ~12 MB traffic)
// Bandwidth-bound at 23.3 TB/s -> keep fp32, use V_WMMA_F32_16X16X4_F32.
//
// Layouts tuned so every WMMA operand pair is one aligned b64 load:
//   - LDS activation tiles [x][ic]  (ic innermost, padded for bank spread)
//   - LDS weight tiles     [n][k]   (k innermost, padded)
//   - out1 / V stored planar [b][ch][HW]  (contiguous along p for k_gram)
// ---------------------------------------------------------------------------
#define B_   4
#define C_   32
#define H_   64
#define W_   64
#define HW_  4096
#define OC_  8
#define NEG_SLOPE 0.01f

typedef __attribute__((ext_vector_type(2))) float v2f;
typedef __attribute__((ext_vector_type(4))) float v4f;
typedef __attribute__((ext_vector_type(8))) float v8f;

// f32 WMMA 16x16x4:  D(16x16,f32) = A(16x4,f32) * B(4x16,f32) + C
// A frag: lane<16 -> (M=lane,   K=k0..k0+1); lane>=16 -> (M=lane-16, K=k0+2..k0+3)
// B frag: lane<16 -> (K=k0..k0+1, N=lane);   lane>=16 -> (K=k0+2..k0+3, N=lane-16)
// C/D  : VGPR r -> M=r (lanes 0-15) / M=r+8 (lanes 16-31), N = lane%16
__device__ __forceinline__ v8f wmma_f32(v2f a, v2f b, v8f c) {
  return __builtin_amdgcn_wmma_f32_16x16x4_f32(
      false, a, false, b, (short)0, c, false, false);
}

__device__ __forceinline__ float leaky(float v) {
  return v >= 0.f ? v : NEG_SLOPE * v;
}

// ---------------------------------------------------------------------------
// Kernel 1: out1 = leaky(conv3x3(cat,w1)+b1) -> planar [B][8][HW]
// Block = one image row (b,y); 4 waves = 4 x-tiles of 16.
// K order tap-major: k = (ky*3+kx)*64 + ic.
// ---------------------------------------------------------------------------
__global__ void k_conv1(const float* __restrict__ in1, const float* __restrict__ in2,
                        const float* __restrict__ w1, const float* __restrict__ b1,
                        float* __restrict__ out1) {
  __shared__ float Bls[16 * 580];      // [n][k], k padded 576->580 (37 KB)
  __shared__ float Act[3 * 66 * 68];   // [ky][x(-1..64)][ic pad 64->68] (54 KB)
  __shared__ float bias[8];

  const int tid = threadIdx.x;
  const int blk = blockIdx.x;          // b*64 + y
  const int b = blk >> 6, y = blk & 63;

  // Weights -> [n][k], tap-major k = tap*64+ic, zero-padded n=8..15.
  for (int k = tid; k < 576; k += 128) {
    const int tap = k >> 6, ic = k & 63;
#pragma unroll
    for (int n = 0; n < 16; ++n)
      Bls[n * 580 + k] = (n < OC_) ? w1[n * 576 + ic * 9 + tap] : 0.f;
  }
  if (tid < 8) bias[tid] = b1[tid];

  // Activations with halo; coalesced reads (xp inner), transposed LDS write.
  for (int i = tid; i < 3 * 64 * 66; i += 128) {
    const int t = i / 66;              // t = ky*64 + ic
    const int xp = i - t * 66;         // 0..65 -> x = xp-1
    const int ky = t >> 6, ic = t & 63;
    const int yy = y + ky - 1, xx = xp - 1;
    float v = 0.f;
    if ((unsigned)yy < (unsigned)H_ && (unsigned)xx < (unsigned)W_) {
      const float* src = (ic < C_) ? in1 : in2;
      v = src[((b * C_ + (ic & 31)) << 12) + (yy << 6) + xx];
    }
    Act[(ky * 66 + xp) * 68 + ic] = v;
  }
  // gfx1250 global_prefetch_b8 hint for the next row's data.
  if (y + 2 < H_)
    __builtin_prefetch(&in1[(b * C_ << 12) + ((y + 2) << 6) + (tid & 63)], 0, 1);
  __syncthreads();

  const int lane = tid & 31, half = lane >> 4, lm = lane & 15;
  const int x0 = (tid >> 5) << 4;

  v8f acc = {};
#pragma unroll
  for (int ky = 0; ky < 3; ++ky) {
#pragma unroll
    for (int kx = 0; kx < 3; ++kx) {
      const int tap = ky * 3 + kx;
      const float* ap = &Act[(ky * 66 + x0 + lm + kx) * 68 + half * 2];
      const float* bp = &Bls[lm * 580 + tap * 64 + half * 2];
      for (int icb = 0; icb < 64; icb += 4) {    // 16 WMMA / tap
        v2f a  = *(const v2f*)ap;                // one b64 LDS load
        v2f bf = *(const v2f*)bp;                // one b64 LDS load
        acc = wmma_f32(a, bf, acc);
        ap += 4; bp += 4;
      }
    }
  }

  const int p0 = (y << 6) + x0;
  if (lm < OC_) {
    const float bb = bias[lm];
    v4f lo, hi;
#pragma unroll
    for (int r = 0; r < 4; ++r) lo[r] = leaky(acc[r] + bb);
#pragma unroll
    for (int r = 0; r < 4; ++r) hi[r] = leaky(acc[r + 4] + bb);
    float* dst = &out1[(b * OC_ + lm) * HW_ + p0 + half * 8];
    *(v4f*)dst = lo;                             // two b128 stores
    *(v4f*)(dst + 4) = hi;
  }
}

// ---------------------------------------------------------------------------
// Kernel 2: V = leaky(conv3x3(out1,w2)+b2) + (conv1x1(cat,w11)+b11)
// -> planar [B][8][HW].  Two accumulator chains.
// ---------------------------------------------------------------------------
__global__ void k_conv2(const float* __restrict__ in1, const float* __restrict__ in2,
                        const float* __restrict__ out1,
                        const float* __restrict__ w2, const float* __restrict__ b2,
                        const float* __restrict__ w11, const float* __restrict__ b11,
                        float* __restrict__ V) {
  __shared__ float Bls2[16 * 76];      // [n][k], k = tap*8+ic, pad 72->76
  __shared__ float Bls3[16 * 68];      // [n][k], k = ic,       pad 64->68
  __shared__ float Act2[3 * 66 * 12];  // [ky][x][ic pad 8->12]
  __shared__ float Act3[64 * 68];      // [x][ic pad 64->68]
  __shared__ float bias2[8], bias3[8];

  const int tid = threadIdx.x;
  const int blk = blockIdx.x;
  const int b = blk >> 6, y = blk & 63;

  if (tid < 72) {
    const int k = tid, tap = k >> 3, ic = k & 7;
#pragma unroll
    for (int n = 0; n < 16; ++n)
      Bls2[n * 76 + k] = (n < OC_) ? w2[n * 72 + ic * 9 + tap] : 0.f;
  }
  if (tid < 64) {
    const int k = tid;
#pragma unroll
    for (int n = 0; n < 16; ++n)
      Bls3[n * 68 + k] = (n < OC_) ? w11[n * 64 + k] : 0.f;
  }
  for (int i = tid; i < 3 * 8 * 66; i += 128) {
    const int t = i / 66;              // t = ky*8 + ic
    const int xp = i - t * 66;
    const int ky = t >> 3, ic = t & 7;
    const int yy = y + ky - 1, xx = xp - 1;
    float v = 0.f;
    if ((unsigned)yy < (unsigned)H_ && (unsigned)xx < (unsigned)W_)
      v = out1[(b * OC_ + ic) * HW_ + (yy << 6) + xx];   // planar, coalesced
    Act2[(ky * 66 + xp) * 12 + ic] = v;
  }
  for (int i = tid; i < 64 * 64; i += 128) {
    const int ic = i >> 6, xx = i & 63;
    const float* src = (ic < C_) ? in1 : in2;
    Act3[xx * 68 + ic] = src[((b * C_ + (ic & 31)) << 12) + (y << 6) + xx];
  }
  if (tid < 8) { bias2[tid] = b2[tid]; bias3[tid] = b11[tid]; }
  __syncthreads();

  const int lane = tid & 31, half = lane >> 4, lm = lane & 15;
  const int x0 = (tid >> 5) << 4;

  v8f acc2 = {};   // conv2 branch (gets leaky)
  v8f acc3 = {};   // 1x1 branch
#pragma unroll
  for (int ky = 0; ky < 3; ++ky) {
#pragma unroll
    for (int kx = 0; kx < 3; ++kx) {
      const int tap = ky * 3 + kx;
      const float* ap = &Act2[(ky * 66 + x0 + lm + kx) * 12 + half * 2];
      const float* bp = &Bls2[lm * 76 + tap * 8 + half * 2];
#pragma unroll
      for (int icb = 0; icb < 8; icb += 4) {     // 2 WMMA / tap
        v2f a  = *(const v2f*)ap;
        v2f bf = *(const v2f*)bp;
        acc2 = wmma_f32(a, bf, acc2);
        ap += 4; bp += 4;
      }
    }
  }
  {
    const float* ap = &Act3[(x0 + lm) * 68 + half * 2];
    const float* bp = &Bls3[lm * 68 + half * 2];
    for (int icb = 0; icb < 64; icb += 4) {      // 16 WMMA
      v2f a  = *(const v2f*)ap;
      v2f bf = *(const v2f*)bp;
      acc3 = wmma_f32(a, bf, acc3);
      ap += 4; bp += 4;
    }
  }

  const int p0 = (y << 6) + x0;
  if (lm < OC_) {
    const float bb2 = bias2[lm], bb3 = bias3[lm];
    v4f lo, hi;
#pragma unroll
    for (int r = 0; r < 4; ++r) lo[r] = leaky(acc2[r] + bb2) + (acc3[r] + bb3);
#pragma unroll
    for (int r = 0; r < 4; ++r) hi[r] = leaky(acc2[r + 4] + bb2) + (acc3[r + 4] + bb3);
    float* dst = &V[(b * OC_ + lm) * HW_ + p0 + half * 8];
    *(v4f*)dst = lo;
    *(v4f*)(dst + 4) = hi;
  }
}

// ---------------------------------------------------------------------------
// Kernel 3: Wout[b] = V^T * [ V | X1 ]  ->  [8 x 40] per batch.
// Planar V makes both operands contiguous along p: 2 b64 loads + 1 WMMA/iter.
// Padded rows/cols read a zeroed scratch strip (no masks, no divergence).
// ---------------------------------------------------------------------------
__global__ void k_gram(const float* __restrict__ V, const float* __restrict__ in1,
                       const float* __restrict__ zpad, float* __restrict__ Wout) {
  const int wave = blockIdx.x * (blockDim.x >> 5) + (threadIdx.x >> 5);
  const int lane = threadIdx.x & 31;
  const int half = lane >> 4;
  const int lm   = lane & 15;
  const int kchunk = wave & 3;          // 4 K-chunks of 1024
  const int nt     = (wave >> 2) % 3;   // 3 N-tiles of 16 (40 real cols)
  const int b      = wave / 12;
  const int ncol   = nt * 16 + lm;

  const int pbeg = kchunk * 1024 + half * 2;

  // A row i = lm: V plane lm, or the zero strip for padded rows 8..15.
  const float* ap = (lm < OC_) ? V + (b * OC_ + lm) * HW_ + pbeg : zpad + pbeg;

  // B col: [0..7]=V planes, [8..39]=input1 planes, else zero strip.
  const float* bp;
  if (ncol < 8)       bp = V + (b * OC_ + ncol) * HW_ + pbeg;
  else if (ncol < 40) bp = in1 + (b * C_ + (ncol - 8)) * HW_ + pbeg;
  else                bp = zpad + pbeg;

  v8f acc = {};
#pragma unroll 4
  for (int k0 = 0; k0 < 1024; k0 += 4) {
    v2f a  = *(const v2f*)ap;           // one b64 global load
    v2f bf = *(const v2f*)bp;           // one b64 global load
    acc = wmma_f32(a, bf, acc);
    ap += 4; bp += 4;
  }

  if (ncol < 40 && half == 0) {         // rows 8..15 of D are zero padding
#pragma unroll
    for (int r = 0; r < 8; ++r)
      atomicAdd(&Wout[(b * 8 + r) * 40 + ncol], acc[r]);
  }
}

// ---------------------------------------------------------------------------
// Kernel 4: per-batch Gauss-Jordan on [M | T] (8x40) -> U = M^-1 T (8x32).
// M = V^T V is SPD -> no pivoting. One block per batch.
// ---------------------------------------------------------------------------
__global__ void k_solve(const float* __restrict__ Wout, float* __restrict__ U) {
  __shared__ float A[8][40];
  __shared__ float f[8];
  const int b = blockIdx.x;
  for (int i = threadIdx.x; i < 320; i += blockDim.x)
    A[i / 40][i % 40] = Wout[b * 320 + i];
  __syncthreads();

  for (int k = 0; k < 8; ++k) {
    const float piv = A[k][k];
    __syncthreads();
    if (threadIdx.x < 40) A[k][threadIdx.x] *= (1.f / piv);
    __syncthreads();
    if (threadIdx.x < 8) f[threadIdx.x] = A[threadIdx.x][k];
    __syncthreads();
    for (int i = threadIdx.x; i < 320; i += blockDim.x) {
      const int r = i / 40, cq = i % 40;
      if (r != k) A[r][cq] -= f[r] * A[k][cq];
    }
    __syncthreads();
  }
  for (int i = threadIdx.x; i < 256; i += blockDim.x)
    U[b * 256 + i] = A[i / 32][8 + (i % 32)];
}

// ---------------------------------------------------------------------------
// Kernel 5: Y = V @ U -> NCHW output.  K=8 (2 WMMA); b128 epilogue stores.
// ---------------------------------------------------------------------------
__global__ void k_apply(const float* __restrict__ V, const float* __restrict__ U,
                        float* __restrict__ out) {
  const int wave = blockIdx.x * (blockDim.x >> 5) + (threadIdx.x >> 5);
  const int lane = threadIdx.x & 31;
  const int half = lane >> 4;
  const int lm   = lane & 15;
  const int nt   = wave & 1;            // 2 tiles cover 32 channels
  const int t    = wave >> 1;
  const int b    = t >> 8;
  const int p0   = (t & 255) << 4;
  const int ncol = nt * 16 + lm;

  const float* vp = V + (b * OC_) * HW_ + p0 + lm;   // A[m][k] = V[k][p0+m]
  const float* up = U + (b * 8) * 32 + ncol;

  v8f acc = {};
#pragma unroll
  for (int k0 = 0; k0 < 8; k0 += 4) {
    const int kb = k0 + half * 2;
    v2f a, bf;
    a.x  = vp[kb * HW_];
    a.y  = vp[(kb + 1) * HW_];
    bf.x = up[kb * 32];
    bf.y = up[(kb + 1) * 32];
    acc = wmma_f32(a, bf, acc);
  }

  v4f lo, hi;
#pragma unroll
  for (int r = 0; r < 4; ++r) { lo[r] = acc[r]; hi[r] = acc[r + 4]; }
  float* dst = &out[(b * C_ + ncol) * HW_ + p0 + half * 8];
  *(v4f*)dst = lo;                      // NCHW, two b128 stores
  *(v4f*)(dst + 4) = hi;
}

__global__ void k_zero(float* __restrict__ p, int n) {
  int i = blockIdx.x * blockDim.x + threadIdx.x;
  if (i < n) p[i] = 0.f;
}

// ---------------------------------------------------------------------------
// Launch: d_in order = input1, input2, w1, b1, w2, b2, w11, b11
// Workspace (floats):
//   out1 planar [0      .. 131071]
//   V    planar [131072 .. 262143]
//   Wout        [262144 .. 263423]  (zeroed each call)
//   U           [263424 .. 264447]
//   zpad        [264448 .. 268551]  (zeroed each call)
// ---------------------------------------------------------------------------
extern "C" void kernel_launch(void* const* d_in, const int* in_sizes, int n_in,
                              void* d_out, int out_size, void* d_ws, size_t ws_size,
                              hipStream_t stream) {
  const float* in1 = (const float*)d_in[0];
  const float* in2 = (const float*)d_in[1];
  const float* w1  = (const float*)d_in[2];
  const float* b1  = (const float*)d_in[3];
  const float* w2  = (const float*)d_in[4];
  const float* b2  = (const float*)d_in[5];
  const float* w11 = (const float*)d_in[6];
  const float* b11 = (const float*)d_in[7];
  float* out = (float*)d_out;

  float* ws   = (float*)d_ws;
  float* out1 = ws;
  float* V    = ws + 131072;
  float* Wout = ws + 262144;
  float* U    = ws + 263424;
  float* zpad = ws + 264448;

  // Zero Wout + U + zpad strip (contiguous 6408 floats) every call.
  k_zero<<<26, 256, 0, stream>>>(Wout, 6408);

  // One block per image row: B*H = 256 blocks, 4 waves each.
  k_conv1<<<B_ * H_, 128, 0, stream>>>(in1, in2, w1, b1, out1);
  k_conv2<<<B_ * H_, 128, 0, stream>>>(in1, in2, out1, w2, b2, w11, b11, V);

  // 48 waves = 12 blocks x 4 waves
  k_gram<<<12, 128, 0, stream>>>(V, in1, zpad, Wout);

  k_solve<<<B_, 64, 0, stream>>>(Wout, U);

  // 2048 waves = 512 blocks x 4 waves
  k_apply<<<512, 128, 0, stream>>>(V, U, out);
}